// HistPredictor_704374637368
// MI455X (gfx1250) — compile-verified
//
#include <hip/hip_runtime.h>
#include <hip/hip_bf16.h>

// GMM -> 128x128 histogram splat, batch-per-block, WMMA quadratic form.
// B=256 batches, K=16 mixtures, G=128*128 grid, range [-2,2]^2.
//
// D-tile (V_WMMA_F32_16X16X4_F32, M=mixture, N=grid point, K=8 padded) directly
// yields log2( coef_k * exp(-.5*mahal) ) because log2(e) and log2(coef) are
// folded into the A-matrix rows:
//   A[k,:] = L2E * [-.5 ia, -ib, -.5 ic, ia*mu+ib*mv, ib*mu+ic*mv, -.5*q(mu)] + [0,...,0,log2 coef]
//   B[:,g] = [gu^2, gu*gv, gv^2, gu, gv, 1, 0, 0]
// so hist contribution = raw v_exp_f32 of D.  Lanes 0-15 accumulate rows 0..7,
// lanes 16-31 rows 8..15 into two LDS partial histograms; halves are summed in
// the normalization/write-out pass (no cross-lane ops in the hot loop).

typedef float v2f __attribute__((ext_vector_type(2)));
typedef float v8f __attribute__((ext_vector_type(8)));

#define GMM_B 256
#define GMM_K 16
#define GRID_N 128
#define G_TOT (GRID_N * GRID_N)        // 16384
#define WAVES 8
#define ITERS GRID_N                   // 128 u-rows per wave

// Raw v_exp_f32 (no denorm-range guard): args < -126 flush to 0, which is the
// numerically correct histogram contribution anyway (< 1e-38).
static __device__ __forceinline__ float fast_exp2(float x) {
#if __has_builtin(__builtin_amdgcn_exp2f)
    return __builtin_amdgcn_exp2f(x);
#else
    return exp2f(x);
#endif
}

__global__ void __launch_bounds__(256) gmm_hist_wmma_kernel(
    const float* __restrict__ mu,     // [B,K,2]
    const float* __restrict__ sigma,  // [B,K,2]
    const float* __restrict__ cov12,  // [B,K]
    const float* __restrict__ pi_,    // [B,K]
    float* __restrict__ out)          // [B,128,128]
{
    extern __shared__ float smem[];   // [16384] lo-half + [16384] hi-half + [8] sums
    float* histLo = smem;
    float* histHi = smem + G_TOT;
    float* wsum   = smem + 2 * G_TOT;

    const int b    = blockIdx.x;
    const int tid  = threadIdx.x;
    const int lane = tid & 31;
    const int w    = tid >> 5;
    const bool lo  = (lane < 16);
    const int k    = lane & 15;       // mixture row M / grid column N (per half-wave)

    // ---- per-mixture parameters (lane owns mixture k = lane%16) ----
    const int bk = b * GMM_K + k;
    const float mu_u = mu[bk * 2 + 0];
    const float mu_v = mu[bk * 2 + 1];
    const float s_u  = fmaxf(sigma[bk * 2 + 0], 0.001f);
    const float s_v  = fmaxf(sigma[bk * 2 + 1], 0.001f);
    const float off  = cov12[bk];
    const float wpi  = pi_[bk];

    const float su2 = s_u * s_u;
    const float sv2 = s_v * s_v;
    const float c11 = su2 + 1e-6f;
    const float c22 = sv2 + 1e-6f;
    const float det_full = c11 * c22 - off * off;
    const bool valid = (det_full > 0.0f) && (det_full == det_full);
    const float det_safe = valid ? det_full : 1.0f;
    const float ia = valid ? (c22 / det_safe) : (1.0f / su2);
    const float ib = valid ? (-off / det_safe) : 0.0f;
    const float ic = valid ? (c11 / det_safe) : (1.0f / sv2);
    const float det = valid ? det_full : (su2 * sv2);
    const float coef = wpi / (6.28318530717958647692f * sqrtf(det + 1e-6f));

    // A-row coefficients: -0.5*mahal pre-scaled by log2(e); log2(coef) folded
    // into the constant term so exp2(D) is the full weighted contribution.
    const float L2E = 1.44269504088896340736f;
    const float p0 = L2E * (-0.5f * ia);
    const float p1 = L2E * (-ib);
    const float p2 = L2E * (-0.5f * ic);
    const float p3 = L2E * (ia * mu_u + ib * mu_v);
    const float p4 = L2E * (ib * mu_u + ic * mu_v);
    const float p5 = L2E * (-0.5f * (ia * mu_u * mu_u + 2.0f * ib * mu_u * mu_v
                                     + ic * mu_v * mu_v))
                   + __log2f(coef);

    // A-matrix 16x4 f32 layout: lanes 0-15 hold M=lane, {v0,v1}={K0,K1};
    // lanes 16-31 hold M=lane-16, {v0,v1}={K2,K3}.  Two K-chunks cover K=0..7.
    v2f a0, a1;
    a0.x = lo ? p0 : p2;  a0.y = lo ? p1 : p3;   // K {0,1} | {2,3}
    a1.x = lo ? p4 : 0.f; a1.y = lo ? p5 : 0.f;  // K {4,5} | {6,7}

    // ---- wave-invariant grid geometry ----
    // tile t = w + 8*it  ->  u-row i = it, v-column j = 16*w + k (fixed strip)
    const float scale = 4.0f / 127.0f;           // linspace(-2,2,128) step
    const int   j     = (w << 4) + k;
    const float gv    = -2.0f + (float)j * scale;
    const float gv2   = gv * gv;

    // B-matrix 4x16 f32: lane holds column N=lane%16;
    // lanes 0-15: {v0,v1}={K0,K1}; lanes 16-31: {v0,v1}={K2,K3}.
    v2f b1;                                      // chunk1 (K 4..7) is loop-invariant
    b1.x = lo ? gv  : 0.f;
    b1.y = lo ? 1.f : 0.f;

    float* histPart = (lo ? histLo : histHi) + j;   // + 128*it per iteration
    float waveSum = 0.0f;

#pragma unroll 2
    for (int it = 0; it < ITERS; ++it) {
        const float gu = -2.0f + (float)it * scale;

        v2f b0;
        b0.x = lo ? gu * gu : gv2;
        b0.y = lo ? gu * gv : gu;

        v8f c = {};
        c = __builtin_amdgcn_wmma_f32_16x16x4_f32(false, a0, false, b0,
                                                  (short)0, c, false, false);
        c = __builtin_amdgcn_wmma_f32_16x16x4_f32(false, a1, false, b1,
                                                  (short)0, c, false, false);

        // D layout: VGPR r, lane l -> M = r + 8*(l>=16), N = l%16.
        // Sum this half-wave's 8 mixture rows for grid point (it, j).
        float h = 0.0f;
#pragma unroll
        for (int r = 0; r < 8; ++r)
            h += fast_exp2(c[r]);

        histPart[it * GRID_N] = h;     // unconditional: lo/hi halves disjoint
        waveSum += h;
    }

    // full-wave sum (lo rows + hi rows = all 16 mixtures over the strip)
#pragma unroll
    for (int o = 16; o >= 1; o >>= 1)
        waveSum += __shfl_xor(waveSum, o, 32);
    if (lane == 0) wsum[w] = waveSum;
    __syncthreads();

    float total = 0.0f;
#pragma unroll
    for (int x = 0; x < WAVES; ++x) total += wsum[x];
    const float inv = (total > 0.0f) ? (1.0f / total) : 1.0f;

    // merge halves + normalize + coalesced float4 write-out
    const float4* hLo4 = (const float4*)histLo;
    const float4* hHi4 = (const float4*)histHi;
    float4* o4 = (float4*)out + (size_t)b * (G_TOT / 4);
#pragma unroll
    for (int u = 0; u < G_TOT / 4 / 256; ++u) {
        const int idx = u * 256 + tid;
        const float4 vl = hLo4[idx];
        const float4 vh = hHi4[idx];
        float4 v;
        v.x = (vl.x + vh.x) * inv;
        v.y = (vl.y + vh.y) * inv;
        v.z = (vl.z + vh.z) * inv;
        v.w = (vl.w + vh.w) * inv;
        o4[idx] = v;
    }
}

extern "C" void kernel_launch(void* const* d_in, const int* in_sizes, int n_in,
                              void* d_out, int out_size, void* d_ws, size_t ws_size,
                              hipStream_t stream) {
    (void)in_sizes; (void)n_in; (void)out_size; (void)d_ws; (void)ws_size;
    const float* mu    = (const float*)d_in[0];
    const float* sigma = (const float*)d_in[1];
    const float* cov12 = (const float*)d_in[2];
    const float* pi_   = (const float*)d_in[3];
    float* out = (float*)d_out;

    const size_t shmem = (2 * G_TOT + WAVES) * sizeof(float);  // 128 KB + sums
    dim3 grid(GMM_B), block(256);
    gmm_hist_wmma_kernel<<<grid, block, shmem, stream>>>(mu, sigma, cov12, pi_, out);
}